// Convolve_67053029425400
// MI455X (gfx1250) — compile-verified
//
#include <hip/hip_runtime.h>
#include <math.h>

typedef __attribute__((ext_vector_type(2))) float v2f;
typedef __attribute__((ext_vector_type(8))) float v8f;

namespace {
constexpr int B_ = 8, N_ = 8192, C_ = 256, H_ = 256, O_ = 256, K_ = 64, CH_ = 512;
constexpr float ALPHA = 0.3f;
constexpr float EPSF  = 1e-6f;

// Workspace layout (float offsets)
constexpr int WS_ACC  = 0;            // [B_*H_]  weighted-sum accumulator (pre-division)
constexpr int WS_WBUF = 2048;         // [K_]     gathered neighbor weights
constexpr int WS_WSUM = 2112;         // [1]      sum of neighbor weights
constexpr int WS_HID  = 2116;         // [B_*O_]  hidden_new before row-normalize
}

__device__ __forceinline__ float leaky(float x) { return x >= 0.f ? x : ALPHA * x; }

// ---------------------------------------------------------------------------
// Kernel 0: zero accumulator, gather neighbor weights, serial-sum them
// (serial sum by one lane keeps the result bit-deterministic across replays)
// ---------------------------------------------------------------------------
__global__ void k_init(const float* __restrict__ weights,
                       const int* __restrict__ nbr,
                       const int* __restrict__ node_id_p,
                       float* __restrict__ ws) {
  const int tid = threadIdx.x;
  for (int i = tid; i < B_ * H_; i += blockDim.x) ws[WS_ACC + i] = 0.f;
  if (tid < K_) {
    const int nb = nbr[tid];
    ws[WS_WBUF + tid] = weights[(size_t)nb * N_ + node_id_p[0]];
  }
  __syncthreads();
  if (tid == 0) {
    float s = 0.f;
    for (int k = 0; k < K_; ++k) s += ws[WS_WBUF + k];
    ws[WS_WSUM] = s;
  }
}

// ---------------------------------------------------------------------------
// Kernel 1: neighbor_hiddens = leaky(gather(emb) @ Qw + Qb), then weighted
// accumulation over neighbors into ws[WS_ACC].
// One wave computes one 16x16 output tile with V_WMMA_F32_16X16X4_F32,
// stepping K by 4 (64 WMMAs per tile).
// ---------------------------------------------------------------------------
__global__ void k_gemm1(const float* __restrict__ emb,
                        const float* __restrict__ Qw,
                        const float* __restrict__ Qb,
                        const int* __restrict__ nbr,
                        float* __restrict__ ws) {
  const int wave = blockIdx.x * (blockDim.x >> 5) + (threadIdx.x >> 5);
  const int lane = threadIdx.x & 31;
  const int half = lane >> 4;            // 0 -> K+0/K+1, 1 -> K+2/K+3 (ISA A/B f32 layout)
  const int l16  = lane & 15;
  const int tile_m = wave >> 4;          // 0..31  (512 rows / 16)
  const int tile_n = wave & 15;          // 0..15  (256 cols / 16)

  // A operand: row m of the gathered neighbor matrix
  const int m = tile_m * 16 + l16;       // 0..511
  const int b = m >> 6;                  // batch index (64 neighbors per batch)
  const int k = m & 63;                  // neighbor slot
  const float* __restrict__ arow = emb + ((size_t)b * N_ + (size_t)nbr[k]) * C_;
  const int n = tile_n * 16 + l16;       // output column (B operand lane = column)

  v8f acc = {};
  for (int kk = 0; kk < C_; kk += 4) {
    const int ka = kk + half * 2;
    v2f a, bv;
    a.x  = arow[ka];
    a.y  = arow[ka + 1];
    bv.x = Qw[(size_t)ka * H_ + n];
    bv.y = Qw[(size_t)(ka + 1) * H_ + n];
    acc = __builtin_amdgcn_wmma_f32_16x16x4_f32(false, a, false, bv,
                                                (short)0, acc, false, false);
  }

  // Epilogue: lane holds rows M = v + 8*half, column n of the D tile.
  // Whole 16-row tile lies in a single batch (64 % 16 == 0).
  const float qb = Qb[n];
  float s = 0.f;
#pragma unroll
  for (int v = 0; v < 8; ++v) {
    const int rm = tile_m * 16 + v + 8 * half;
    const int rk = rm & 63;
    s += leaky(acc[v] + qb) * ws[WS_WBUF + rk];
  }
  const int rb = (tile_m * 16) >> 6;
  atomicAdd(&ws[WS_ACC + rb * H_ + n], s);
}

// ---------------------------------------------------------------------------
// Kernel 2: hidden_new = leaky(concat(node_emb, weighted_sum) @ Ww + Wb)
// Single 16-row M tile (rows 8..15 zero padded) x 16 N tiles = 16 waves.
// 512-deep K -> 128 WMMAs per wave.
// ---------------------------------------------------------------------------
__global__ void k_gemm2(const float* __restrict__ emb,
                        const float* __restrict__ Ww,
                        const float* __restrict__ Wb,
                        const int* __restrict__ node_id_p,
                        float* __restrict__ ws) {
  const int wave = threadIdx.x >> 5;     // 0..15 -> N tile
  const int lane = threadIdx.x & 31;
  const int half = lane >> 4;
  const int l16  = lane & 15;
  const int n = wave * 16 + l16;
  const int node = node_id_p[0];

  const int m = l16;                     // A row; valid when < B_
  const bool rowok = (m < B_);
  const float* __restrict__ nrow = emb + ((size_t)(rowok ? m : 0) * N_ + node) * C_;
  const float invw = 1.f / (ws[WS_WSUM] + EPSF);
  const float* __restrict__ accp = ws + WS_ACC + (rowok ? m : 0) * H_;

  v8f acc = {};
  for (int kk = 0; kk < CH_; kk += 4) {
    const int ka = kk + half * 2;        // ka is even; pairs never straddle C_
    v2f a, bv;
    const float a0 = (ka     < C_) ? nrow[ka]     : accp[ka - C_]     * invw;
    const float a1 = (ka + 1 < C_) ? nrow[ka + 1] : accp[ka + 1 - C_] * invw;
    a.x  = rowok ? a0 : 0.f;
    a.y  = rowok ? a1 : 0.f;
    bv.x = Ww[(size_t)ka * O_ + n];
    bv.y = Ww[(size_t)(ka + 1) * O_ + n];
    acc = __builtin_amdgcn_wmma_f32_16x16x4_f32(false, a, false, bv,
                                                (short)0, acc, false, false);
  }

  const float wb = Wb[n];
#pragma unroll
  for (int v = 0; v < 8; ++v) {
    const int M = v + 8 * half;
    if (M < B_) ws[WS_HID + M * O_ + n] = leaky(acc[v] + wb);
  }
}

// ---------------------------------------------------------------------------
// Kernel 3: row L2-normalize: out[b] = h[b] / (||h[b]|| + eps)
// One block per row, LDS tree reduction of squares.
// ---------------------------------------------------------------------------
__global__ void k_norm(const float* __restrict__ ws, float* __restrict__ out) {
  __shared__ float red[256];
  const int b = blockIdx.x;
  const int t = threadIdx.x;
  const float x = ws[WS_HID + b * O_ + t];
  red[t] = x * x;
  __syncthreads();
  for (int s = 128; s > 0; s >>= 1) {
    if (t < s) red[t] += red[t + s];
    __syncthreads();
  }
  const float nrm = sqrtf(red[0]);
  out[b * O_ + t] = x / (nrm + EPSF);
}

// ---------------------------------------------------------------------------
extern "C" void kernel_launch(void* const* d_in, const int* in_sizes, int n_in,
                              void* d_out, int out_size, void* d_ws, size_t ws_size,
                              hipStream_t stream) {
  const float* emb     = (const float*)d_in[0];
  const float* weights = (const float*)d_in[1];
  const float* Qw      = (const float*)d_in[2];
  const float* Qb      = (const float*)d_in[3];
  const float* Ww      = (const float*)d_in[4];
  const float* Wb      = (const float*)d_in[5];
  const int*   nbr     = (const int*)d_in[6];
  const int*   node    = (const int*)d_in[7];
  float* ws  = (float*)d_ws;
  float* out = (float*)d_out;

  k_init <<<1,  256, 0, stream>>>(weights, nbr, node, ws);
  k_gemm1<<<64, 256, 0, stream>>>(emb, Qw, Qb, nbr, ws);
  k_gemm2<<<1,  512, 0, stream>>>(emb, Ww, Wb, node, ws);
  k_norm <<<B_, 256, 0, stream>>>(ws, out);
}